// ElementaryBlock_12111807774818
// MI455X (gfx1250) — compile-verified
//
#include <hip/hip_runtime.h>

typedef __attribute__((ext_vector_type(2))) float vf2;
typedef __attribute__((ext_vector_type(8))) float vf8;

#define NB    16
#define NPTS  2048
#define D     64
#define E     128   // 2*D
#define KNBR  16
#define NOUT  128   // DOUT + NM

// ---------------------------------------------------------------- zero z region
__global__ void zero_z(float* __restrict__ out_z) {
    int t = blockIdx.x * blockDim.x + threadIdx.x;
    if (t < NB * 64) out_z[t] = 0.0f;
}

// ---------------------------------------------------------------- sq norms (wave per point)
__global__ void sqnorm_kernel(const float* __restrict__ x, float* __restrict__ sq) {
    int wid  = (blockIdx.x * blockDim.x + threadIdx.x) >> 5;   // point id 0..32767
    int lane = threadIdx.x & 31;
    const float* p = x + (size_t)wid * D + lane * 2;
    float a = p[0], b = p[1];
    float s = a * a + b * b;
    #pragma unroll
    for (int o = 16; o > 0; o >>= 1) s += __shfl_xor(s, o, 32);
    if (lane == 0) sq[wid] = s;
}

// ---------------------------------------------------------------- zadd[b][c] = bias + z@W row terms
__global__ void zadd_kernel(const float* __restrict__ z,
                            const float* __restrict__ Wmz, const float* __restrict__ bmz,
                            const float* __restrict__ Wvz, const float* __restrict__ bvz,
                            const float* __restrict__ bmx, const float* __restrict__ bvx,
                            float* __restrict__ zadd) {
    int t = blockIdx.x * blockDim.x + threadIdx.x;
    if (t >= NB * NOUT) return;
    int b = t >> 7, c = t & 127;
    const float* zb = z + b * 64;
    float acc;
    if (c < 64) {
        acc = bmz[c] + bmx[c];
        for (int i = 0; i < 64; ++i) acc += zb[i] * Wmz[i * 64 + c];
    } else {
        int c2 = c - 64;
        acc = bvz[c2] + bvx[c2];
        for (int i = 0; i < 64; ++i) acc += zb[i] * Wvz[i * 64 + c2];
    }
    zadd[t] = acc;
}

// ---------------------------------------------------------------- KNN: 16 rows per block, WMMA dots + top-17
__global__ __launch_bounds__(128) void knn_kernel(const float* __restrict__ x,
                                                  const float* __restrict__ sq,
                                                  int* __restrict__ nbr) {
    __shared__ float tileLDS[4][256];        // per-wave private 16x16 dot tile
    __shared__ float keyLDS[4][16][17];
    __shared__ int   idxLDS[4][16][17];

    const int wave  = threadIdx.x >> 5;
    const int lane  = threadIdx.x & 31;
    const int hlane = lane & 15;
    const int koff  = (lane < 16) ? 0 : 2;
    const int b     = blockIdx.x >> 7;            // 128 row tiles per batch
    const int row0  = (blockIdx.x & 127) << 4;

    const float* xb  = x  + (size_t)b * NPTS * D;
    const float* sqb = sq + b * NPTS;

    // A fragments: x_tile rows (M) x K chunks of 4 (ISA 16x4 fp32 A layout)
    vf2 afr[16];
    {
        const float* ap = xb + (size_t)(row0 + hlane) * D + koff;
        #pragma unroll
        for (int s = 0; s < 16; ++s) afr[s] = *(const vf2*)(ap + 4 * s);
    }
    float sqRow = (lane < 16) ? sqb[row0 + lane] : 0.0f;

    // per-lane (lanes 0..15) sorted top-17 for its row over this wave's columns
    float bk[17]; int bi[17];
    #pragma unroll
    for (int t = 0; t < 17; ++t) { bk[t] = 3.0e38f; bi[t] = 0x7fffffff; }

    const vf8 vzero = {0.f,0.f,0.f,0.f,0.f,0.f,0.f,0.f};

    for (int it = 0; it < 32; ++it) {
        const int col0 = (wave * 32 + it) << 4;
        const float* bp = xb + (size_t)(col0 + hlane) * D + koff;
        if (it < 31)  // prefetch next column tile (global_prefetch_b8)
            __builtin_prefetch(xb + (size_t)(col0 + 16 + hlane) * D + koff, 0, 3);

        // preload all 16 B fragments so loads overlap, then chain the WMMAs
        vf2 bfr[16];
        #pragma unroll
        for (int s = 0; s < 16; ++s) bfr[s] = *(const vf2*)(bp + 4 * s);

        vf8 acc = vzero;
        #pragma unroll
        for (int s = 0; s < 16; ++s)
            acc = __builtin_amdgcn_wmma_f32_16x16x4_f32(false, afr[s], false, bfr[s],
                                                        (short)0, acc, false, false);

        // spill 16x16 dot tile: VGPR r holds rows r (lanes 0-15) / r+8 (lanes 16-31)
        // tile slab is per-wave private; DS ops are in-order per wave -> no barrier needed
        #pragma unroll
        for (int r = 0; r < 8; ++r) {
            int m = (lane < 16) ? r : (r + 8);
            tileLDS[wave][m * 16 + hlane] = acc[r];
        }
        if (lane < 16) {
            const float* sqc = sqb + col0;
            #pragma unroll 4
            for (int j = 0; j < 16; ++j) {
                float dot = tileLDS[wave][lane * 16 + j];
                float key = fmaxf(sqRow + sqc[j] - 2.0f * dot, 0.0f);
                int   idx = col0 + j;
                if ((key < bk[16]) || (key == bk[16] && idx < bi[16])) {
                    float ck = key; int ci = idx;
                    #pragma unroll
                    for (int p = 0; p <= 16; ++p) {   // bubble insert, static indices only
                        bool bet = (ck < bk[p]) || (ck == bk[p] && ci < bi[p]);
                        float tk = bet ? bk[p] : ck;  int ti = bet ? bi[p] : ci;
                        bk[p]    = bet ? ck    : bk[p]; bi[p] = bet ? ci : bi[p];
                        ck = tk; ci = ti;
                    }
                }
            }
        }
    }

    if (lane < 16) {
        #pragma unroll
        for (int t = 0; t < 17; ++t) { keyLDS[wave][lane][t] = bk[t]; idxLDS[wave][lane][t] = bi[t]; }
    }
    __syncthreads();   // only cross-wave sync: before the 4-way merge

    // 4-way merge of per-wave top-17 lists; drop entry 0 (self)
    if (threadIdx.x < 16) {
        const int row = threadIdx.x;
        int p0 = 0, p1 = 0, p2 = 0, p3 = 0;
        int* dst = nbr + (size_t)(b * NPTS + row0 + row) * KNBR;
        for (int t = 0; t < 17; ++t) {
            float k0 = (p0 < 17) ? keyLDS[0][row][p0] : 3.1e38f; int i0 = (p0 < 17) ? idxLDS[0][row][p0] : 0x7fffffff;
            float k1 = (p1 < 17) ? keyLDS[1][row][p1] : 3.1e38f; int i1 = (p1 < 17) ? idxLDS[1][row][p1] : 0x7fffffff;
            float k2 = (p2 < 17) ? keyLDS[2][row][p2] : 3.1e38f; int i2 = (p2 < 17) ? idxLDS[2][row][p2] : 0x7fffffff;
            float k3 = (p3 < 17) ? keyLDS[3][row][p3] : 3.1e38f; int i3 = (p3 < 17) ? idxLDS[3][row][p3] : 0x7fffffff;
            float kq = k0; int iq = i0; int w = 0;
            if (k1 < kq || (k1 == kq && i1 < iq)) { kq = k1; iq = i1; w = 1; }
            if (k2 < kq || (k2 == kq && i2 < iq)) { kq = k2; iq = i2; w = 2; }
            if (k3 < kq || (k3 == kq && i3 < iq)) { kq = k3; iq = i3; w = 3; }
            if      (w == 0) ++p0;
            else if (w == 1) ++p1;
            else if (w == 2) ++p2;
            else             ++p3;
            if (t > 0) dst[t - 1] = iq;
        }
    }
}

// ---------------------------------------------------------------- pairs(16x128) @ W(128x128), 4 points per wave
__global__ __launch_bounds__(256) void pair_gemm_kernel(const float* __restrict__ x,
                                                        const int* __restrict__ nbr,
                                                        const float* __restrict__ Wmx,
                                                        const float* __restrict__ Wvx,
                                                        const float* __restrict__ zadd,
                                                        float* __restrict__ out_x,
                                                        float* __restrict__ out_z) {
    __shared__ vf2 Wlds[64 * NOUT];   // 64 KB: W rows (k,k+1) pair-interleaved -> 1 ds_load_b64 per B frag

    const int wave  = threadIdx.x >> 5;
    const int lane  = threadIdx.x & 31;
    const int hlane = lane & 15;
    const int koff  = (lane < 16) ? 0 : 2;
    const int kp    = (lane < 16) ? 0 : 1;   // k-pair selector

    // stage W = [Wmx | Wvx]: Wlds[p*128+n] = {W[2p][n], W[2p+1][n]}
    for (int i = threadIdx.x; i < 64 * NOUT; i += 256) {
        int p = i >> 7, n = i & 127;
        const float* src = (n < 64) ? (Wmx + n) : (Wvx + (n - 64));
        vf2 w; w.x = src[(2 * p) * 64]; w.y = src[(2 * p + 1) * 64];
        Wlds[i] = w;
    }
    __syncthreads();

    const int gbase = blockIdx.x * 32 + wave * 4;  // 32 points per block; b uniform (32 | 2048)
    const int b     = gbase >> 11;
    const float* za = zadd + b * NOUT;

    float zsum[4] = {0.f, 0.f, 0.f, 0.f};
    const vf8 vzero = {0.f,0.f,0.f,0.f,0.f,0.f,0.f,0.f};

    #pragma unroll 1
    for (int pp = 0; pp < 4; ++pp) {
        const int g = gbase + pp;
        const int nrow = nbr[(size_t)g * KNBR + hlane];     // this lane's neighbor row
        const float* xself = x + (size_t)g * D;
        const float* xnb   = x + ((size_t)b * NPTS + nrow) * D;

        // A fragments: 32 K-chunks of 4 (first 16 = self half, last 16 = gathered neighbor half)
        vf2 afr[32];
        #pragma unroll
        for (int s = 0; s < 16; ++s) afr[s]      = *(const vf2*)(xself + 4 * s + koff);
        #pragma unroll
        for (int s = 0; s < 16; ++s) afr[16 + s] = *(const vf2*)(xnb   + 4 * s + koff);

        vf8 acc[8];
        #pragma unroll
        for (int t = 0; t < 8; ++t) acc[t] = vzero;

        #pragma unroll
        for (int s = 0; s < 32; ++s) {
            const vf2* wr = &Wlds[(2 * s + kp) * NOUT + hlane];
            vf2 bfs[8];
            #pragma unroll
            for (int t = 0; t < 8; ++t) bfs[t] = wr[t * 16];
            #pragma unroll
            for (int t = 0; t < 8; ++t)
                acc[t] = __builtin_amdgcn_wmma_f32_16x16x4_f32(false, afr[s], false, bfs[t],
                                                               (short)0, acc[t], false, false);
        }

        // epilogue: ReLU(C + zadd), sum over the 16 neighbor rows
        #pragma unroll
        for (int t = 0; t < 8; ++t) {
            const int col = t * 16 + hlane;
            const float add = za[col];
            float sum = 0.0f;
            #pragma unroll
            for (int r = 0; r < 8; ++r) sum += fmaxf(acc[t][r] + add, 0.0f);
            sum += __shfl_xor(sum, 16, 32);           // combine rows r and r+8 halves (wave32)
            if (t < 4) {
                if (lane < 16) out_x[(size_t)g * 64 + col] = sum * (1.0f / 16.0f);
            } else {
                zsum[t - 4] += sum;                   // defer z atomics, accumulate over 4 points
            }
        }
    }

    // one atomic per lane per tile for all 4 points
    #pragma unroll
    for (int t = 0; t < 4; ++t) {
        if (lane < 16)
            atomicAdd(&out_z[b * 64 + t * 16 + hlane], zsum[t] * (1.0f / (16.0f * 2048.0f)));
    }
}

// ---------------------------------------------------------------- launcher
extern "C" void kernel_launch(void* const* d_in, const int* in_sizes, int n_in,
                              void* d_out, int out_size, void* d_ws, size_t ws_size,
                              hipStream_t stream) {
    const float* x   = (const float*)d_in[0];
    const float* z   = (const float*)d_in[1];
    const float* Wmx = (const float*)d_in[2];
    const float* bmx = (const float*)d_in[3];
    const float* Wvx = (const float*)d_in[4];
    const float* bvx = (const float*)d_in[5];
    const float* Wmz = (const float*)d_in[6];
    const float* bmz = (const float*)d_in[7];
    const float* Wvz = (const float*)d_in[8];
    const float* bvz = (const float*)d_in[9];

    float* out_x = (float*)d_out;
    float* out_z = out_x + (size_t)NB * NPTS * 64;

    float* sq   = (float*)d_ws;                 // 16*2048 floats
    float* zadd = sq + NB * NPTS;               // 16*128 floats
    int*   nbr  = (int*)(zadd + NB * NOUT);     // 16*2048*16 ints

    zero_z       <<<4,    256, 0, stream>>>(out_z);
    sqnorm_kernel<<<4096, 256, 0, stream>>>(x, sq);
    zadd_kernel  <<<8,    256, 0, stream>>>(z, Wmz, bmz, Wvz, bvz, bmx, bvx, zadd);
    knn_kernel   <<<NB * (NPTS / 16), 128, 0, stream>>>(x, sq, nbr);
    pair_gemm_kernel<<<(NB * NPTS) / 32, 256, 0, stream>>>(x, nbr, Wmx, Wvx, zadd, out_x, out_z);
}